// RGCN_7138235646653
// MI455X (gfx1250) — compile-verified
//
#include <hip/hip_runtime.h>
#include <hip/hip_bf16.h>

constexpr int NN = 50000;
constexpr int EE = 800000;
constexpr int RR = 8;
constexpr int BB = 4;
constexpr int HH = 128;
constexpr int OO = 16;

typedef float v2f __attribute__((ext_vector_type(2)));
typedef float v8f __attribute__((ext_vector_type(8)));

__global__ void rgcn_zero_kernel(uint32_t* __restrict__ p, long n) {
  long i = (long)blockIdx.x * blockDim.x + threadIdx.x;
  long stride = (long)gridDim.x * blockDim.x;
  for (; i < n; i += stride) p[i] = 0u;
}

__global__ void rgcn_count_kernel(const int* __restrict__ rel, int* __restrict__ counts) {
  int e = blockIdx.x * blockDim.x + threadIdx.x;
  if (e < EE) atomicAdd(&counts[rel[e]], 1);
}

// Build W1 TRANSPOSED: W1t[r'][o][h'] so the WMMA B-operand pair (k0,k0+1) for a
// fixed output column is contiguous (enables b64 loads in the hot loop).
// Torch-faithful scramble: flat combine row j = r'*HH + h'; i = j>>3, r = j&7.
__global__ void rgcn_w1_build_kernel(const float* __restrict__ w_comp1,
                                     const float* __restrict__ bases1,
                                     float* __restrict__ W1t) {
  int t = blockIdx.x * blockDim.x + threadIdx.x; // [0, RR*HH*OO)
  if (t >= RR * HH * OO) return;
  int row = t >> 4;         // j = r'*HH + h' in [0,1024)
  int o   = t & 15;
  int i = row >> 3;
  int r = row & 7;
  float s = 0.f;
#pragma unroll
  for (int b = 0; b < BB; ++b)
    s += w_comp1[r * BB + b] * bases1[((size_t)b * HH + i) * OO + o];
  const int rp = row >> 7;   // r' = row / HH
  const int hp = row & 127;  // h' = row % HH
  W1t[((size_t)rp * OO + o) * HH + hp] = s;
}

__global__ void rgcn_scan_kernel(const int* __restrict__ counts,
                                 int* __restrict__ offsets,
                                 int* __restrict__ tileStart,
                                 int* __restrict__ cursors) {
  if (threadIdx.x != 0 || blockIdx.x != 0) return;
  int off = 0, ts = 0;
  for (int r = 0; r < RR; ++r) {
    offsets[r]   = off;
    tileStart[r] = ts;
    cursors[r]   = off;
    off += counts[r];
    ts  += (counts[r] + 15) >> 4;
  }
  offsets[RR]   = off;
  tileStart[RR] = ts;
}

__global__ void rgcn_fill_kernel(const int* __restrict__ rel,
                                 int* __restrict__ cursors,
                                 int* __restrict__ bucket) {
  int e = blockIdx.x * blockDim.x + threadIdx.x;
  if (e >= EE) return;
  int pos = atomicAdd(&cursors[rel[e]], 1);
  bucket[pos] = e;
}

// Layer 0: one wave per edge; lane handles 4 consecutive h (float4 of bases0 rows)
__global__ __launch_bounds__(256)
void rgcn_layer0_kernel(const int* __restrict__ src, const int* __restrict__ dst,
                        const int* __restrict__ rel, const float* __restrict__ ew,
                        const float* __restrict__ bases0, const float* __restrict__ w_comp0,
                        float* __restrict__ h0) {
  const int wave = threadIdx.x >> 5;
  const int lane = threadIdx.x & 31;
  const int e = blockIdx.x * 8 + wave;
  if (e >= EE) return;
  const int idx = rel[e] * NN + src[e];   // torch-faithful fused index
  const int i = idx >> 3;                 // /R
  const int r = idx & 7;                  // %R
  const float c0 = w_comp0[r * BB + 0];
  const float c1 = w_comp0[r * BB + 1];
  const float c2 = w_comp0[r * BB + 2];
  const float c3 = w_comp0[r * BB + 3];
  const float w = ew[e];
  const int h = lane * 4;
  const float4 v0 = *(const float4*)(bases0 + ((size_t)0 * NN + i) * HH + h);
  const float4 v1 = *(const float4*)(bases0 + ((size_t)1 * NN + i) * HH + h);
  const float4 v2 = *(const float4*)(bases0 + ((size_t)2 * NN + i) * HH + h);
  const float4 v3 = *(const float4*)(bases0 + ((size_t)3 * NN + i) * HH + h);
  float4 a;
  a.x = w * (c0 * v0.x + c1 * v1.x + c2 * v2.x + c3 * v3.x);
  a.y = w * (c0 * v0.y + c1 * v1.y + c2 * v2.y + c3 * v3.y);
  a.z = w * (c0 * v0.z + c1 * v1.z + c2 * v2.z + c3 * v3.z);
  a.w = w * (c0 * v0.w + c1 * v1.w + c2 * v2.w + c3 * v3.w);
  float* o = h0 + (size_t)dst[e] * HH + h;
  atomicAdd(o + 0, a.x);
  atomicAdd(o + 1, a.y);
  atomicAdd(o + 2, a.z);
  atomicAdd(o + 3, a.w);
}

// Layer 1: one wave per tile of 16 same-relation edges.
// D(16 edges x 16 outs) = [ew * relu(h0[src])](16x128) @ W1[r](128x16)
// via 32 chained V_WMMA_F32_16X16X4_F32. B fed from transposed W1t -> b64 loads.
__global__ __launch_bounds__(32)
void rgcn_layer1_wmma_kernel(const int* __restrict__ bucket,
                             const int* __restrict__ offsets,
                             const int* __restrict__ tileStart,
                             const int* __restrict__ counts,
                             const int* __restrict__ src,
                             const int* __restrict__ dst,
                             const float* __restrict__ ew,
                             const float* __restrict__ h0,
                             const float* __restrict__ W1t,
                             float* __restrict__ outAcc) {
  const int b = blockIdx.x;
  if (b >= tileStart[RR]) return;  // wave-uniform exit
  int r = 0;
  while (r < RR - 1 && b >= tileStart[r + 1]) ++r;
  const int tile = b - tileStart[r];
  const int cnt  = counts[r];
  const int base = offsets[r] + tile * 16;

  const int lane = threadIdx.x;  // 0..31, one wave
  const int m    = lane & 15;    // A-row / D-column index
  const int half = lane >> 4;    // K sub-offset for A/B (16x4 f32 layout)

  // A row for edge slot m (zero-fill invalid tail slots; keeps EXEC all-1s at WMMA)
  const int s_m = tile * 16 + m;
  const float* rowp = h0;        // safe pointer
  float ewm = 0.f;
  if (s_m < cnt) {
    const int e = bucket[base + m];
    rowp = h0 + (size_t)src[e] * HH;
    ewm  = ew[e];
  }

  const int n = m;  // output column handled by this lane
  const float* w1p = W1t + ((size_t)r * OO + n) * HH;  // contiguous in k

  v8f c = {0.f, 0.f, 0.f, 0.f, 0.f, 0.f, 0.f, 0.f};
#pragma unroll
  for (int s = 0; s < HH / 4; ++s) {
    const int k0 = 4 * s + 2 * half;   // even -> 8B aligned b64 loads
    const v2f ain = *(const v2f*)(rowp + k0);
    float a0 = ain[0] > 0.f ? ain[0] : 0.f;  // fused relu
    float a1 = ain[1] > 0.f ? ain[1] : 0.f;
    v2f av;
    av[0] = a0 * ewm;                  // fused edge-weight
    av[1] = a1 * ewm;
    const v2f bv = *(const v2f*)(w1p + k0);
    c = __builtin_amdgcn_wmma_f32_16x16x4_f32(
        /*neg_a=*/false, av, /*neg_b=*/false, bv,
        /*c_mod=*/(short)0, c, /*reuse_a=*/false, /*reuse_b=*/false);
  }

  // D layout: c[j] = D[j + 8*half][n]; scatter per edge slot
#pragma unroll
  for (int j = 0; j < 8; ++j) {
    const int mp = j + 8 * half;
    const int sp = tile * 16 + mp;
    if (sp < cnt) {
      const int e2 = bucket[base + mp];
      atomicAdd(&outAcc[(size_t)dst[e2] * OO + n], c[j]);
    }
  }
}

__global__ void rgcn_softmax_kernel(const float* __restrict__ acc, float* __restrict__ out) {
  int nd = blockIdx.x * blockDim.x + threadIdx.x;
  if (nd >= NN) return;
  const float* p = acc + (size_t)nd * OO;
  float v[OO];
  float mx = p[0];
#pragma unroll
  for (int k = 0; k < OO; ++k) { v[k] = p[k]; mx = v[k] > mx ? v[k] : mx; }
  float s = 0.f;
#pragma unroll
  for (int k = 0; k < OO; ++k) { v[k] = expf(v[k] - mx); s += v[k]; }
  const float inv = 1.f / s;
  float* q = out + (size_t)nd * OO;
#pragma unroll
  for (int k = 0; k < OO; ++k) q[k] = v[k] * inv;
}

extern "C" void kernel_launch(void* const* d_in, const int* in_sizes, int n_in,
                              void* d_out, int out_size, void* d_ws, size_t ws_size,
                              hipStream_t stream) {
  (void)in_sizes; (void)n_in; (void)out_size; (void)ws_size;
  const int*   src      = (const int*)d_in[0];
  const int*   dst      = (const int*)d_in[1];
  const int*   rel      = (const int*)d_in[2];
  const float* ew       = (const float*)d_in[3];
  const float* bases0   = (const float*)d_in[4];
  const float* bases1   = (const float*)d_in[5];
  const float* w_comp0  = (const float*)d_in[6];
  const float* w_comp1  = (const float*)d_in[7];
  float* out = (float*)d_out;

  // Workspace layout (contiguous f32 region first, then ints)
  float* h0     = (float*)d_ws;                 // NN*HH
  float* W1t    = h0 + (size_t)NN * HH;         // RR*OO*HH (transposed)
  float* outAcc = W1t + (size_t)RR * HH * OO;   // NN*OO
  int* counts    = (int*)(outAcc + (size_t)NN * OO); // 8
  int* offsets   = counts + RR;                 // 9
  int* tileStart = offsets + (RR + 1);          // 9
  int* cursors   = tileStart + (RR + 1);        // 8
  int* bucket    = cursors + RR;                // EE

  // Zero h0, W1t, outAcc, counts (contiguous words from ws start)
  const long zeroWords = (long)NN * HH + (long)RR * HH * OO + (long)NN * OO + RR;
  rgcn_zero_kernel<<<2048, 256, 0, stream>>>((uint32_t*)d_ws, zeroWords);

  rgcn_count_kernel<<<(EE + 255) / 256, 256, 0, stream>>>(rel, counts);
  rgcn_w1_build_kernel<<<(RR * HH * OO + 255) / 256, 256, 0, stream>>>(w_comp1, bases1, W1t);
  rgcn_scan_kernel<<<1, 1, 0, stream>>>(counts, offsets, tileStart, cursors);
  rgcn_fill_kernel<<<(EE + 255) / 256, 256, 0, stream>>>(rel, cursors, bucket);

  rgcn_layer0_kernel<<<(EE + 7) / 8, 256, 0, stream>>>(src, dst, rel, ew, bases0, w_comp0, h0);

  const int maxTiles = EE / 16 + RR;  // upper bound; blocks past real tile count exit
  rgcn_layer1_wmma_kernel<<<maxTiles, 32, 0, stream>>>(bucket, offsets, tileStart, counts,
                                                       src, dst, ew, h0, W1t, outAcc);

  rgcn_softmax_kernel<<<(NN + 255) / 256, 256, 0, stream>>>(outAcc, out);
}